// Seq2SeqAttention_38019050505016
// MI455X (gfx1250) — compile-verified
//
#include <hip/hip_runtime.h>

// Problem constants (from reference setup_inputs)
#define HH  256   // hidden
#define TT  512   // sequence length
#define BB  256   // batch
#define DD  32    // input dim
#define AA  128   // attention dim
#define HOR 24
#define NQ  3
#define BT  16    // batch tile per workgroup (== WMMA M)
#define PL  132   // LDS plane stride (floats): 16B-aligned, low-conflict
#define WROWS 1024  // 4*HH gate rows

typedef __attribute__((ext_vector_type(2))) float v2f;
typedef __attribute__((ext_vector_type(4))) float v4f;
typedef __attribute__((ext_vector_type(8))) float v8f;

__device__ __forceinline__ v8f wmma4(v2f a, v2f b, v8f c) {
  // D = A(16x4 f32) * B(4x16 f32) + C(16x16 f32), full fp32 WMMA
  return __builtin_amdgcn_wmma_f32_16x16x4_f32(
      false, a, false, b, (short)0, c, false, false);
}

__device__ __forceinline__ v8f splat8(float v) {
  v8f r;
#pragma unroll
  for (int i = 0; i < 8; ++i) r[i] = v;
  return r;
}

#if __has_builtin(__builtin_amdgcn_tanhf)
__device__ __forceinline__ float ftanh(float x) {
  return __builtin_amdgcn_tanhf(x);
}
#else
__device__ __forceinline__ float ftanh(float x) {
  float cl = fminf(fmaxf(x, -15.0f), 15.0f);
  float e = __expf(2.0f * cl);
  return (e - 1.0f) * __builtin_amdgcn_rcpf(e + 1.0f);
}
#endif
__device__ __forceinline__ float fsig(float x) {
  return fmaf(0.5f, ftanh(0.5f * x), 0.5f);  // exact identity
}

// split-plane index: j -> (plane s, pos).  Plane s holds k = {2s,2s+1} mod 4.
__device__ __forceinline__ int plane_idx(int m, int j) {
  int s = (j >> 1) & 1;
  int pos = ((j >> 2) << 1) | (j & 1);
  return (m * 2 + s) * PL + pos;
}

// ---------------------------------------------------------------------------
// Repack a (1024 x 256) row-major weight matrix into per-row split planes:
// P[(n*2+s)*128 + (k/4)*2 + (k%2)] = W[n][k], s = (k%4)>=2.
// A lane's B-operand for two k-steps becomes one 16-byte load.
// ---------------------------------------------------------------------------
__global__ __launch_bounds__(256) void repack_kernel(
    const float* __restrict__ W, float* __restrict__ P) {
  int idx = blockIdx.x * 256 + threadIdx.x;  // over 1024*256
  int n = idx >> 8;
  int k = idx & 255;
  int s = (k >> 1) & 1;
  int pos = ((k >> 2) << 1) | (k & 1);
  P[((size_t)((n << 1) | s)) * 128 + pos] = W[idx];
}

// ---------------------------------------------------------------------------
// Encoder: 2-layer LSTM over T=512 steps. 16 workgroups x 512 threads.
// Wave w owns hidden cols [16w,16w+16); gate tiles i/f/g/o share (m,j) per
// lane slot, so the cell update runs entirely in registers; c never leaves
// VGPRs, h round-trips through split-plane LDS (b128 A loads).
// ---------------------------------------------------------------------------
__global__ __launch_bounds__(512, 1) void enc_kernel(
    const float* __restrict__ x,
    const float* __restrict__ Wih0, const float* __restrict__ b0,
    const float* __restrict__ Pwhh0, const float* __restrict__ Pwih1,
    const float* __restrict__ Pwhh1, const float* __restrict__ b1,
    float* __restrict__ enc) {
  __shared__ __align__(16) float h0p[BT * 2 * PL];
  __shared__ __align__(16) float h1p[BT * 2 * PL];
  const int tid   = threadIdx.x;
  const int lane  = tid & 31;
  const int wave  = tid >> 5;
  const int bbase = blockIdx.x * BT;
  const int mrow  = lane & 15;   // A row / B col within tile
  const int sl    = lane >> 4;   // K-slot half (plane select)
  const int khi   = sl << 1;
  const int hi8   = sl << 3;     // C row offset (0 or 8)
  const int j0    = wave << 4;   // this wave's hidden-col slice

  for (int i = tid; i < BT * 2 * PL; i += 512) {
    h0p[i] = 0.0f;
    h1p[i] = 0.0f;
  }
  v8f c0r = splat8(0.0f);
  v8f c1r = splat8(0.0f);

  const float* pw_hh0[4];
  const float* pw_ih1[4];
  const float* pw_hh1[4];
  const float* pw_ih0[4];
  float bias0[4], bias1[4];
#pragma unroll
  for (int g = 0; g < 4; ++g) {
    int n = g * 256 + j0 + mrow;  // gate row / B column for this lane
    pw_hh0[g] = Pwhh0 + ((size_t)(n * 2 + sl)) * 128;
    pw_ih1[g] = Pwih1 + ((size_t)(n * 2 + sl)) * 128;
    pw_hh1[g] = Pwhh1 + ((size_t)(n * 2 + sl)) * 128;
    pw_ih0[g] = Wih0 + (size_t)n * DD + khi;  // small, loop-invariant
    bias0[g] = b0[n];
    bias1[g] = b1[n];
  }
  const float* ha0 = &h0p[(mrow * 2 + sl) * PL];
  const float* ha1 = &h1p[(mrow * 2 + sl) * PL];
  const int jw   = j0 + mrow;              // hidden col this lane updates
  const int sw   = (jw >> 1) & 1;
  const int posw = ((jw >> 2) << 1) | (jw & 1);
  __syncthreads();

  for (int t = 0; t < TT; ++t) {
    // ------------- layer 0: gates = x_t@Wih0^T + h0@Whh0^T + b0 -----------
    v8f acc[4];
#pragma unroll
    for (int g = 0; g < 4; ++g) acc[g] = splat8(bias0[g]);

    const float* xrow = x + ((size_t)(bbase + mrow) * TT + t) * DD + khi;
    v2f ax[8];
#pragma unroll
    for (int kk = 0; kk < 8; ++kk) {
      ax[kk].x = xrow[kk * 4];
      ax[kk].y = xrow[kk * 4 + 1];
    }
#pragma unroll
    for (int kk = 0; kk < 8; ++kk) {
#pragma unroll
      for (int g = 0; g < 4; ++g) {
        v2f bm;
        bm.x = pw_ih0[g][kk * 4];
        bm.y = pw_ih0[g][kk * 4 + 1];
        acc[g] = wmma4(ax[kk], bm, acc[g]);
      }
    }
#pragma unroll 2
    for (int p = 0; p < HH / 8; ++p) {  // each p = two k-steps
      v4f a4 = *(const v4f*)(ha0 + (p << 2));
      v4f w0 = *(const v4f*)(pw_hh0[0] + (p << 2));
      v4f w1 = *(const v4f*)(pw_hh0[1] + (p << 2));
      v4f w2 = *(const v4f*)(pw_hh0[2] + (p << 2));
      v4f w3 = *(const v4f*)(pw_hh0[3] + (p << 2));
      v2f alo = a4.xy, ahi = a4.zw;
      acc[0] = wmma4(alo, w0.xy, acc[0]);
      acc[1] = wmma4(alo, w1.xy, acc[1]);
      acc[2] = wmma4(alo, w2.xy, acc[2]);
      acc[3] = wmma4(alo, w3.xy, acc[3]);
      acc[0] = wmma4(ahi, w0.zw, acc[0]);
      acc[1] = wmma4(ahi, w1.zw, acc[1]);
      acc[2] = wmma4(ahi, w2.zw, acc[2]);
      acc[3] = wmma4(ahi, w3.zw, acc[3]);
    }
    __syncthreads();  // all waves done reading h0_old
#pragma unroll
    for (int v = 0; v < 8; ++v) {
      float cn = fsig(acc[1][v]) * c0r[v] + fsig(acc[0][v]) * ftanh(acc[2][v]);
      float hn = fsig(acc[3][v]) * ftanh(cn);
      c0r[v] = cn;
      h0p[((v + hi8) * 2 + sw) * PL + posw] = hn;
    }
    __syncthreads();  // h0_new visible

    // ------------- layer 1: gates = h0n@Wih1^T + h1@Whh1^T + b1 -----------
#pragma unroll
    for (int g = 0; g < 4; ++g) acc[g] = splat8(bias1[g]);
#pragma unroll 2
    for (int p = 0; p < HH / 8; ++p) {
      v4f a4 = *(const v4f*)(ha0 + (p << 2));
      v4f w0 = *(const v4f*)(pw_ih1[0] + (p << 2));
      v4f w1 = *(const v4f*)(pw_ih1[1] + (p << 2));
      v4f w2 = *(const v4f*)(pw_ih1[2] + (p << 2));
      v4f w3 = *(const v4f*)(pw_ih1[3] + (p << 2));
      v2f alo = a4.xy, ahi = a4.zw;
      acc[0] = wmma4(alo, w0.xy, acc[0]);
      acc[1] = wmma4(alo, w1.xy, acc[1]);
      acc[2] = wmma4(alo, w2.xy, acc[2]);
      acc[3] = wmma4(alo, w3.xy, acc[3]);
      acc[0] = wmma4(ahi, w0.zw, acc[0]);
      acc[1] = wmma4(ahi, w1.zw, acc[1]);
      acc[2] = wmma4(ahi, w2.zw, acc[2]);
      acc[3] = wmma4(ahi, w3.zw, acc[3]);
    }
#pragma unroll 2
    for (int p = 0; p < HH / 8; ++p) {
      v4f a4 = *(const v4f*)(ha1 + (p << 2));
      v4f w0 = *(const v4f*)(pw_hh1[0] + (p << 2));
      v4f w1 = *(const v4f*)(pw_hh1[1] + (p << 2));
      v4f w2 = *(const v4f*)(pw_hh1[2] + (p << 2));
      v4f w3 = *(const v4f*)(pw_hh1[3] + (p << 2));
      v2f alo = a4.xy, ahi = a4.zw;
      acc[0] = wmma4(alo, w0.xy, acc[0]);
      acc[1] = wmma4(alo, w1.xy, acc[1]);
      acc[2] = wmma4(alo, w2.xy, acc[2]);
      acc[3] = wmma4(alo, w3.xy, acc[3]);
      acc[0] = wmma4(ahi, w0.zw, acc[0]);
      acc[1] = wmma4(ahi, w1.zw, acc[1]);
      acc[2] = wmma4(ahi, w2.zw, acc[2]);
      acc[3] = wmma4(ahi, w3.zw, acc[3]);
    }
    __syncthreads();
#pragma unroll
    for (int v = 0; v < 8; ++v) {
      float cn = fsig(acc[1][v]) * c1r[v] + fsig(acc[0][v]) * ftanh(acc[2][v]);
      float hn = fsig(acc[3][v]) * ftanh(cn);
      c1r[v] = cn;
      h1p[((v + hi8) * 2 + sw) * PL + posw] = hn;
    }
    __syncthreads();
    // coalesced write of this step's layer-1 output to enc[b][t][:]
    for (int i = tid; i < BT * HH; i += 512) {
      int m = i >> 8, j = i & 255;
      enc[((size_t)(bbase + m) * TT + t) * HH + j] = h1p[plane_idx(m, j)];
    }
  }
}

// ---------------------------------------------------------------------------
// Attention energy: energy[b,t] = tanh(enc[b,t,:] @ aW) @ aV.  One wave per
// (b, 4 consecutive t): aW row loaded once (b128/lane) and applied to 4 rows,
// cutting aW L2 traffic 4x.  Lane l owns attention cols [4l,4l+4).
// ---------------------------------------------------------------------------
__global__ __launch_bounds__(256) void energy_kernel(
    const float* __restrict__ enc, const float* __restrict__ aW,
    const float* __restrict__ aV, float* __restrict__ energy) {
  int gw   = blockIdx.x * 8 + (threadIdx.x >> 5);  // over BB*TT/4 waves
  int lane = threadIdx.x & 31;
  int b  = gw >> 7;          // / (TT/4)
  int t4 = (gw & 127) << 2;  // base t (multiple of 4)
  const float* e = enc + ((size_t)b * TT + t4) * HH;
  const float* wbase = aW + 4 * lane;
  v4f acc0 = {0.f, 0.f, 0.f, 0.f};
  v4f acc1 = acc0, acc2 = acc0, acc3 = acc0;
  for (int h = 0; h < HH; ++h) {
    v4f wr = *(const v4f*)(wbase + (size_t)h * AA);
    acc0 += wr * e[h];
    acc1 += wr * e[h + HH];
    acc2 += wr * e[h + 2 * HH];
    acc3 += wr * e[h + 3 * HH];
  }
  v4f vv = *(const v4f*)(aV + 4 * lane);
  float v0 = ftanh(acc0.x) * vv.x + ftanh(acc0.y) * vv.y +
             ftanh(acc0.z) * vv.z + ftanh(acc0.w) * vv.w;
  float v1 = ftanh(acc1.x) * vv.x + ftanh(acc1.y) * vv.y +
             ftanh(acc1.z) * vv.z + ftanh(acc1.w) * vv.w;
  float v2 = ftanh(acc2.x) * vv.x + ftanh(acc2.y) * vv.y +
             ftanh(acc2.z) * vv.z + ftanh(acc2.w) * vv.w;
  float v3 = ftanh(acc3.x) * vv.x + ftanh(acc3.y) * vv.y +
             ftanh(acc3.z) * vv.z + ftanh(acc3.w) * vv.w;
#pragma unroll
  for (int off = 16; off > 0; off >>= 1) {
    v0 += __shfl_xor(v0, off, 32);
    v1 += __shfl_xor(v1, off, 32);
    v2 += __shfl_xor(v2, off, 32);
    v3 += __shfl_xor(v3, off, 32);
  }
  if (lane == 0) {
    v4f r = {v0, v1, v2, v3};
    *(v4f*)(energy + (size_t)b * TT + t4) = r;
  }
}

// ---------------------------------------------------------------------------
// Softmax over T + context = sum_t w[t]*enc[b,t,:].  One block per batch row.
// ---------------------------------------------------------------------------
__global__ __launch_bounds__(256) void attn_kernel(
    const float* __restrict__ enc, const float* __restrict__ energy,
    float* __restrict__ ctx) {
  __shared__ float w[TT];
  __shared__ float red[256];
  int b = blockIdx.x, tid = threadIdx.x;
  float e0 = energy[(size_t)b * TT + tid];
  float e1 = energy[(size_t)b * TT + tid + 256];
  red[tid] = fmaxf(e0, e1);
  __syncthreads();
  for (int s = 128; s > 0; s >>= 1) {
    if (tid < s) red[tid] = fmaxf(red[tid], red[tid + s]);
    __syncthreads();
  }
  float mx = red[0];
  __syncthreads();
  float x0 = __expf(e0 - mx), x1 = __expf(e1 - mx);
  w[tid] = x0;
  w[tid + 256] = x1;
  red[tid] = x0 + x1;
  __syncthreads();
  for (int s = 128; s > 0; s >>= 1) {
    if (tid < s) red[tid] += red[tid + s];
    __syncthreads();
  }
  float inv = __builtin_amdgcn_rcpf(red[0]);
  float acc = 0.f;
  for (int t = 0; t < TT; ++t)
    acc = fmaf(w[t], enc[((size_t)b * TT + t) * HH + tid], acc);
  ctx[(size_t)b * HH + tid] = acc * inv;
}

// ---------------------------------------------------------------------------
// Decoder: 24-step 2-layer LSTM (input dim 1 folded into accumulator init),
// then head: preds[b,q,o] = last[b,:]@qW[q,o,:] + qb[q,o].
// ---------------------------------------------------------------------------
__global__ __launch_bounds__(512, 1) void dec_kernel(
    const float* __restrict__ ctx, const float* __restrict__ x,
    const float* __restrict__ Wih0, const float* __restrict__ b0,
    const float* __restrict__ Pwhh0, const float* __restrict__ Pwih1,
    const float* __restrict__ Pwhh1, const float* __restrict__ b1,
    const float* __restrict__ qW, const float* __restrict__ qb,
    float* __restrict__ out) {
  __shared__ __align__(16) float h0p[BT * 2 * PL];
  __shared__ __align__(16) float h1p[BT * 2 * PL];
  __shared__ float dinp[BT];
  const int tid   = threadIdx.x;
  const int lane  = tid & 31;
  const int wave  = tid >> 5;
  const int bbase = blockIdx.x * BT;
  const int mrow  = lane & 15;
  const int sl    = lane >> 4;
  const int hi8   = sl << 3;
  const int j0    = wave << 4;

  for (int i = tid; i < BT * HH; i += 512) {
    int m = i >> 8, j = i & 255;
    float cv = ctx[(size_t)(bbase + m) * HH + j];
    int pi = plane_idx(m, j);
    h0p[pi] = cv;
    h1p[pi] = cv;
  }
  v8f c0r, c1r;
#pragma unroll
  for (int v = 0; v < 8; ++v) {
    float cv = ctx[(size_t)(bbase + v + hi8) * HH + j0 + mrow];
    c0r[v] = cv;
    c1r[v] = cv;
  }
  if (tid < BT)
    dinp[tid] = x[((size_t)(bbase + tid) * TT + (TT - 1)) * DD + (DD - 1)];

  const float* pw_hh0[4];
  const float* pw_ih1[4];
  const float* pw_hh1[4];
  float bias0[4], bias1[4], wx0[4];
#pragma unroll
  for (int g = 0; g < 4; ++g) {
    int n = g * 256 + j0 + mrow;
    pw_hh0[g] = Pwhh0 + ((size_t)(n * 2 + sl)) * 128;
    pw_ih1[g] = Pwih1 + ((size_t)(n * 2 + sl)) * 128;
    pw_hh1[g] = Pwhh1 + ((size_t)(n * 2 + sl)) * 128;
    bias0[g] = b0[n];
    bias1[g] = b1[n];
    wx0[g]   = Wih0[n];  // d_Wih0 is (1024,1)
  }
  const float* ha0 = &h0p[(mrow * 2 + sl) * PL];
  const float* ha1 = &h1p[(mrow * 2 + sl) * PL];
  const int jw   = j0 + mrow;
  const int sw   = (jw >> 1) & 1;
  const int posw = ((jw >> 2) << 1) | (jw & 1);
  __syncthreads();

  for (int step = 0; step < HOR; ++step) {
    float di[8];
#pragma unroll
    for (int v = 0; v < 8; ++v) di[v] = dinp[v + hi8];

    // -------- layer 0 --------
    v8f acc[4];
#pragma unroll
    for (int g = 0; g < 4; ++g) {
      acc[g] = splat8(bias0[g]);
#pragma unroll
      for (int v = 0; v < 8; ++v) acc[g][v] = fmaf(di[v], wx0[g], acc[g][v]);
    }
#pragma unroll 2
    for (int p = 0; p < HH / 8; ++p) {
      v4f a4 = *(const v4f*)(ha0 + (p << 2));
      v4f w0 = *(const v4f*)(pw_hh0[0] + (p << 2));
      v4f w1 = *(const v4f*)(pw_hh0[1] + (p << 2));
      v4f w2 = *(const v4f*)(pw_hh0[2] + (p << 2));
      v4f w3 = *(const v4f*)(pw_hh0[3] + (p << 2));
      v2f alo = a4.xy, ahi = a4.zw;
      acc[0] = wmma4(alo, w0.xy, acc[0]);
      acc[1] = wmma4(alo, w1.xy, acc[1]);
      acc[2] = wmma4(alo, w2.xy, acc[2]);
      acc[3] = wmma4(alo, w3.xy, acc[3]);
      acc[0] = wmma4(ahi, w0.zw, acc[0]);
      acc[1] = wmma4(ahi, w1.zw, acc[1]);
      acc[2] = wmma4(ahi, w2.zw, acc[2]);
      acc[3] = wmma4(ahi, w3.zw, acc[3]);
    }
    __syncthreads();
#pragma unroll
    for (int v = 0; v < 8; ++v) {
      float cn = fsig(acc[1][v]) * c0r[v] + fsig(acc[0][v]) * ftanh(acc[2][v]);
      float hn = fsig(acc[3][v]) * ftanh(cn);
      c0r[v] = cn;
      h0p[((v + hi8) * 2 + sw) * PL + posw] = hn;
    }
    __syncthreads();

    // -------- layer 1 --------
#pragma unroll
    for (int g = 0; g < 4; ++g) acc[g] = splat8(bias1[g]);
#pragma unroll 2
    for (int p = 0; p < HH / 8; ++p) {
      v4f a4 = *(const v4f*)(ha0 + (p << 2));
      v4f w0 = *(const v4f*)(pw_ih1[0] + (p << 2));
      v4f w1 = *(const v4f*)(pw_ih1[1] + (p << 2));
      v4f w2 = *(const v4f*)(pw_ih1[2] + (p << 2));
      v4f w3 = *(const v4f*)(pw_ih1[3] + (p << 2));
      v2f alo = a4.xy, ahi = a4.zw;
      acc[0] = wmma4(alo, w0.xy, acc[0]);
      acc[1] = wmma4(alo, w1.xy, acc[1]);
      acc[2] = wmma4(alo, w2.xy, acc[2]);
      acc[3] = wmma4(alo, w3.xy, acc[3]);
      acc[0] = wmma4(ahi, w0.zw, acc[0]);
      acc[1] = wmma4(ahi, w1.zw, acc[1]);
      acc[2] = wmma4(ahi, w2.zw, acc[2]);
      acc[3] = wmma4(ahi, w3.zw, acc[3]);
    }
#pragma unroll 2
    for (int p = 0; p < HH / 8; ++p) {
      v4f a4 = *(const v4f*)(ha1 + (p << 2));
      v4f w0 = *(const v4f*)(pw_hh1[0] + (p << 2));
      v4f w1 = *(const v4f*)(pw_hh1[1] + (p << 2));
      v4f w2 = *(const v4f*)(pw_hh1[2] + (p << 2));
      v4f w3 = *(const v4f*)(pw_hh1[3] + (p << 2));
      v2f alo = a4.xy, ahi = a4.zw;
      acc[0] = wmma4(alo, w0.xy, acc[0]);
      acc[1] = wmma4(alo, w1.xy, acc[1]);
      acc[2] = wmma4(alo, w2.xy, acc[2]);
      acc[3] = wmma4(alo, w3.xy, acc[3]);
      acc[0] = wmma4(ahi, w0.zw, acc[0]);
      acc[1] = wmma4(ahi, w1.zw, acc[1]);
      acc[2] = wmma4(ahi, w2.zw, acc[2]);
      acc[3] = wmma4(ahi, w3.zw, acc[3]);
    }
    __syncthreads();
#pragma unroll
    for (int v = 0; v < 8; ++v) {
      float cn = fsig(acc[1][v]) * c1r[v] + fsig(acc[0][v]) * ftanh(acc[2][v]);
      float hn = fsig(acc[3][v]) * ftanh(cn);
      c1r[v] = cn;
      h1p[((v + hi8) * 2 + sw) * PL + posw] = hn;
      if (j0 == 0 && mrow == 0) dinp[v + hi8] = hn;  // feedback h1n[:, 0]
    }
    __syncthreads();
  }

  // -------- head: preds[b,q,o] = h1[b,:]@qW[q,o,:] + qb[q,o] --------
  for (int idx = tid; idx < BT * NQ * HOR; idx += 512) {
    int m = idx / (NQ * HOR);
    int r = idx % (NQ * HOR);  // r = q*HOR + o
    const float* qr = qW + (size_t)r * HH;
    float s = qb[r];
    for (int h = 0; h < HH; ++h) s = fmaf(h1p[plane_idx(m, h)], qr[h], s);
    out[(size_t)(bbase + m) * (NQ * HOR) + r] = s;
  }
}

// ---------------------------------------------------------------------------
extern "C" void kernel_launch(void* const* d_in, const int* in_sizes, int n_in,
                              void* d_out, int out_size, void* d_ws,
                              size_t ws_size, hipStream_t stream) {
  (void)in_sizes; (void)n_in; (void)out_size; (void)ws_size;
  const float* x     = (const float*)d_in[0];
  const float* eWih0 = (const float*)d_in[1];
  const float* eWhh0 = (const float*)d_in[2];
  const float* eb0   = (const float*)d_in[3];
  const float* eWih1 = (const float*)d_in[4];
  const float* eWhh1 = (const float*)d_in[5];
  const float* eb1   = (const float*)d_in[6];
  const float* dWih0 = (const float*)d_in[7];
  const float* dWhh0 = (const float*)d_in[8];
  const float* db0   = (const float*)d_in[9];
  const float* dWih1 = (const float*)d_in[10];
  const float* dWhh1 = (const float*)d_in[11];
  const float* db1   = (const float*)d_in[12];
  const float* aW    = (const float*)d_in[13];
  const float* aV    = (const float*)d_in[14];
  const float* qW    = (const float*)d_in[15];
  const float* qb    = (const float*)d_in[16];
  float* out = (float*)d_out;

  float* ws     = (float*)d_ws;
  float* enc    = ws;                           // B*T*H floats (128 MB)
  float* energy = enc + (size_t)BB * TT * HH;   // B*T floats
  float* ctx    = energy + (size_t)BB * TT;     // B*H floats
  float* packed = ctx + (size_t)BB * HH;        // 6 x 1024*256 floats
  const size_t WSZ = (size_t)WROWS * HH;
  float* pEwhh0 = packed + 0 * WSZ;
  float* pEwih1 = packed + 1 * WSZ;
  float* pEwhh1 = packed + 2 * WSZ;
  float* pDwhh0 = packed + 3 * WSZ;
  float* pDwih1 = packed + 4 * WSZ;
  float* pDwhh1 = packed + 5 * WSZ;

  const int rgrid = (WROWS * HH) / 256;
  repack_kernel<<<rgrid, 256, 0, stream>>>(eWhh0, pEwhh0);
  repack_kernel<<<rgrid, 256, 0, stream>>>(eWih1, pEwih1);
  repack_kernel<<<rgrid, 256, 0, stream>>>(eWhh1, pEwhh1);
  repack_kernel<<<rgrid, 256, 0, stream>>>(dWhh0, pDwhh0);
  repack_kernel<<<rgrid, 256, 0, stream>>>(dWih1, pDwih1);
  repack_kernel<<<rgrid, 256, 0, stream>>>(dWhh1, pDwhh1);

  enc_kernel<<<BB / BT, 512, 0, stream>>>(x, eWih0, eb0, pEwhh0, pEwih1,
                                          pEwhh1, eb1, enc);
  energy_kernel<<<(BB * TT / 4) / 8, 256, 0, stream>>>(enc, aW, aV, energy);
  attn_kernel<<<BB, 256, 0, stream>>>(enc, energy, ctx);
  dec_kernel<<<BB / BT, 512, 0, stream>>>(ctx, x, dWih0, db0, pDwhh0, pDwih1,
                                          pDwhh1, db1, qW, qb, out);
}